// GCNLayer_3384434229621
// MI455X (gfx1250) — compile-verified
//
#include <hip/hip_runtime.h>

typedef __attribute__((ext_vector_type(2))) float v2f;
typedef __attribute__((ext_vector_type(4))) float v4f;
typedef __attribute__((ext_vector_type(8))) float v8f;

#define N_NODES 8192
#define F       128
#define W1LD    130            // w1 is (128, 130) row-major
#define KSPLIT  8
#define KCHUNK  (N_NODES / KSPLIT)

#define KT          64         // k-values staged per LDS chunk
#define PAIR_STRIDE 1152       // 1024B pair-row +128B pad -> disjoint LDS banks
#define BUF_BYTES   ((KT / 2) * PAIR_STRIDE)   // 36864 B per buffer

// ---------------------------------------------------------------------------
// Phase 2 (split-K x8): stream edges (256MB) + dist (512MB) once from HBM.
//   acc += edges_tile @ xmsg           (v_wmma_f32_16x16x4_f32)
//   s0,s1 += edges*dist                (fused dist_w reduction)
// B (xmsg, pair-interleaved) staged into double-buffered LDS with
// global_load_async_to_lds_b128, consumed via conflict-free ds_load_b64.
// Epilogue: dist_w @ w1d^T as one rank-2 WMMA per tile, then
// global_atomic_add_f32 into out (pre-initialized with x@w2^T by phase 1).
// ---------------------------------------------------------------------------
__global__ __launch_bounds__(256) void gcn_phase2(
    const float* __restrict__ edges, const float* __restrict__ dist,
    const float* __restrict__ xmsg, const float* __restrict__ w1,
    float* __restrict__ out) {
  __shared__ __align__(16) char smem[2 * BUF_BYTES];

  const int lane = threadIdx.x & 31;
  const int wave = threadIdx.x >> 5;
  const int l15  = lane & 15;
  const int kl   = (lane >> 4) * 2;
  const int rowBase = (int)(blockIdx.x & 63) * 128 + wave * 16;
  const int kbeg    = (int)(blockIdx.x >> 6) * KCHUNK;
  const long row    = rowBase + l15;

  // async-stage one KT-chunk of xmsg_pi (linear 32KB) into an LDS buffer
  auto stage = [&](int kc, int parity) {
    const float* src = xmsg + (long)kc * F;            // pair-interleaved, linear
#pragma unroll
    for (int i = 0; i < 8; ++i) {
      const int u = (int)threadIdx.x + i * 256;        // 16-byte unit, 0..2047
      const unsigned ldsa = (unsigned)(size_t)&smem[0] +
                            (unsigned)(parity * BUF_BYTES +
                                       (u >> 6) * PAIR_STRIDE + (u & 63) * 16);
      const float* g = src + u * 4;
      asm volatile("global_load_async_to_lds_b128 %0, %1, off"
                   :: "v"(ldsa), "v"(g) : "memory");
    }
  };

  v8f  acc[8] = {};
  float s0 = 0.f, s1 = 0.f;

  const float* pe = edges + row * N_NODES + kl + kbeg;
  const float* pd = dist + (row * N_NODES + kl) * 2 + 2 * kbeg;

  stage(kbeg, 0);
  asm volatile("s_wait_asynccnt 0x0" ::: "memory");
  __syncthreads();

  const int nchunks = KCHUNK / KT;                     // 16
  for (int c = 0; c < nchunks; ++c) {
    if (c + 1 < nchunks) stage(kbeg + (c + 1) * KT, (c + 1) & 1);

    const char* dsb = smem + (c & 1) * BUF_BYTES +
                      (kl >> 1) * PAIR_STRIDE + l15 * 8;
#pragma unroll 8
    for (int s = 0; s < KT / 4; ++s) {                 // 16 k-steps of K=4
      v2f a = *(const v2f*)pe;                         // edges[row, k..k+1]
      v4f d = *(const v4f*)pd;                         // dist[row, k..k+1, 0..1]
      pe += 4;
      pd += 8;
      if ((s & 1) == 0) {                              // compile-time under unroll
        __builtin_prefetch(pe + 1024, 0, 3);           // 1024 k ahead: 4KB edges, 8KB dist
        __builtin_prefetch(pd + 2048, 0, 3);
      }
      v2f b[8];
#pragma unroll
      for (int t = 0; t < 8; ++t)                      // ds_load_b64, banks 0-31/32-63
        b[t] = *(const v2f*)(dsb + t * 128);
      dsb += 2 * PAIR_STRIDE;
      s0 += a.x * d.x + a.y * d.z;                     // fused dist_w partials
      s1 += a.x * d.y + a.y * d.w;
#pragma unroll
      for (int t = 0; t < 8; ++t)
        acc[t] = __builtin_amdgcn_wmma_f32_16x16x4_f32(false, a, false, b[t],
                                                       (short)0, acc[t], false, false);
    }
    asm volatile("s_wait_asynccnt 0x0" ::: "memory");
    __syncthreads();
  }

  // combine the two lane-half K partials: lane L <-> L+16 hold the same row
  s0 += __shfl_xor(s0, 16, 32);
  s1 += __shfl_xor(s1, 16, 32);

  // rank-2 update via WMMA: A = [dist_w | 0], B rows 0,1 = w1d^T, rows 2,3 = 0
  v2f ad;
  ad.x = (lane < 16) ? s0 : 0.f;
  ad.y = (lane < 16) ? s1 : 0.f;
#pragma unroll
  for (int t = 0; t < 8; ++t) {
    const int col = t * 16 + l15;
    v2f b;
    b.x = (lane < 16) ? w1[col * W1LD + 128] : 0.f;
    b.y = (lane < 16) ? w1[col * W1LD + 129] : 0.f;
    acc[t] = __builtin_amdgcn_wmma_f32_16x16x4_f32(false, ad, false, b,
                                                   (short)0, acc[t], false, false);
  }

  const int rowOff = (lane >> 4) * 8;
#pragma unroll
  for (int t = 0; t < 8; ++t) {
#pragma unroll
    for (int v = 0; v < 8; ++v) {
      const int idx = (rowBase + rowOff + v) * F + t * 16 + l15;
      atomicAdd(&out[idx], acc[t][v]);                 // split-K accumulation
    }
  }
}

// ---------------------------------------------------------------------------
// Phase 1: xmsg_pi = x @ w1[:, :128]^T  (pair-interleaved layout, workspace)
//          out     = x @ w2^T           (split-K accumulation base)
// Pair-interleaved: element (k, col) at ((k>>1)*128 + col)*2 + (k&1), so a
// WMMA B operand {B[k][c], B[k+1][c]} is one contiguous 64-bit load.
// ---------------------------------------------------------------------------
__global__ __launch_bounds__(256) void gcn_phase1(
    const float* __restrict__ x, const float* __restrict__ w1,
    const float* __restrict__ w2, float* __restrict__ xmsg,
    float* __restrict__ out) {
  const int lane = threadIdx.x & 31;
  const int wave = threadIdx.x >> 5;
  const int l15  = lane & 15;
  const int kl   = (lane >> 4) * 2;
  const int rowBase = blockIdx.x * 128 + wave * 16;

  v8f acc1[8] = {};
  v8f acc2[8] = {};

  const float* xptr = x + (rowBase + l15) * F + kl;
  for (int k = 0; k < F; k += 4) {
    v2f a = *(const v2f*)(xptr + k);
    v2f b1[8], b2[8];                       // batch all loads, then all WMMAs
#pragma unroll
    for (int t = 0; t < 8; ++t) {
      const int col = t * 16 + l15;
      b1[t] = *(const v2f*)(w1 + col * W1LD + (k + kl));
      b2[t] = *(const v2f*)(w2 + col * F + (k + kl));
    }
#pragma unroll
    for (int t = 0; t < 8; ++t) {
      acc1[t] = __builtin_amdgcn_wmma_f32_16x16x4_f32(false, a, false, b1[t],
                                                      (short)0, acc1[t], false, false);
      acc2[t] = __builtin_amdgcn_wmma_f32_16x16x4_f32(false, a, false, b2[t],
                                                      (short)0, acc2[t], false, false);
    }
  }

  const int rowOff = (lane >> 4) * 8;       // C layout: lanes16-31 hold M+8
#pragma unroll
  for (int t = 0; t < 8; ++t) {
#pragma unroll
    for (int v = 0; v < 8; ++v) {
      const int row = rowBase + rowOff + v;
      const int col = t * 16 + l15;
      xmsg[((row >> 1) * F + col) * 2 + (row & 1)] = acc1[t][v];
      out[row * F + col] = acc2[t][v];
    }
  }
}

extern "C" void kernel_launch(void* const* d_in, const int* in_sizes, int n_in,
                              void* d_out, int out_size, void* d_ws, size_t ws_size,
                              hipStream_t stream) {
  const float* x     = (const float*)d_in[0];
  const float* edges = (const float*)d_in[1];
  const float* dist  = (const float*)d_in[2];
  const float* w1    = (const float*)d_in[3];
  const float* w2    = (const float*)d_in[4];
  float* out  = (float*)d_out;
  float* xmsg = (float*)d_ws;                          // 8192*128 f32 = 4 MB scratch

  gcn_phase1<<<N_NODES / 128, 256, 0, stream>>>(x, w1, w2, xmsg, out);
  gcn_phase2<<<(N_NODES / 128) * KSPLIT, 256, 0, stream>>>(edges, dist, xmsg, w1, out);
}